// GNN_16999480558341
// MI455X (gfx1250) — compile-verified
//
#include <hip/hip_runtime.h>
#include <hip/hip_bf16.h>

// GCN forward for MI455X (gfx1250, wave32).
// N=100000 nodes, E=1600000 edges, F=128 feats, H=128 hidden, C=2 classes.

typedef float v2f __attribute__((ext_vector_type(2)));
typedef float v8f __attribute__((ext_vector_type(8)));

#define NFEAT 128
#define HIDW  128

// ---------------------------------------------------------------------------
// 1) init: deg[i] = 1.0 (self loop), agg[:] = 0
// ---------------------------------------------------------------------------
__global__ void gcn_init_kernel(float* __restrict__ deg,
                                float* __restrict__ agg,
                                int n, int total) {
    int i = blockIdx.x * blockDim.x + threadIdx.x;
    if (i < total) agg[i] = 0.0f;
    if (i < n)     deg[i] = 1.0f;
}

// ---------------------------------------------------------------------------
// 2) degree: deg[dst[e]] += 1  (global_atomic_add_f32)
// ---------------------------------------------------------------------------
__global__ void gcn_degree_kernel(const int* __restrict__ dst,
                                  float* __restrict__ deg, int nedges) {
    int i = blockIdx.x * blockDim.x + threadIdx.x;
    if (i < nedges) atomicAdd(&deg[dst[i]], 1.0f);
}

// ---------------------------------------------------------------------------
// 3) dinv = rsqrt(deg)
// ---------------------------------------------------------------------------
__global__ void gcn_rsqrt_kernel(float* __restrict__ deg, int n) {
    int i = blockIdx.x * blockDim.x + threadIdx.x;
    if (i < n) {
        float d = deg[i];
        deg[i] = (d > 0.0f) ? rsqrtf(d) : 0.0f;
    }
}

// ---------------------------------------------------------------------------
// 4) xw = x @ W1 using V_WMMA_F32_16X16X4_F32.
//    Block = 128 threads = 4 waves. Block computes a 16x128 output tile;
//    wave w computes columns [w*32, w*32+32) as two 16x16 C tiles.
//    fp32 A 16x4 layout (ISA 7.12.2): lanes 0-15 -> K=(0,1) in (V0,V1),
//    lanes 16-31 -> K=(2,3). So each lane loads a contiguous float2 of x.
//    C/D layout: VGPR r, lanes 0-15 -> M=r, lanes 16-31 -> M=r+8.
//    Store epilogue: wave-uniform full-tile fast path (no per-row branches).
// ---------------------------------------------------------------------------
__global__ void __launch_bounds__(128)
gcn_gemm_xw_kernel(const float* __restrict__ x,
                   const float* __restrict__ W1,
                   float* __restrict__ xw, int nrows) {
    const int wave  = threadIdx.x >> 5;
    const int lane  = threadIdx.x & 31;
    const int tileM = blockIdx.x * 16;
    if (tileM >= nrows) return;                  // block-uniform: EXEC stays full

    const int l15   = lane & 15;
    const int khalf = (lane >> 4) * 2;           // 0 or 2
    int mrow = tileM + l15;
    if (mrow >= nrows) mrow = nrows - 1;         // N % 16 == 0, never triggers

    const int n0    = wave * 32;
    const int ncol0 = n0 + l15;
    const int ncol1 = n0 + 16 + l15;

    v8f c0 = {0.f, 0.f, 0.f, 0.f, 0.f, 0.f, 0.f, 0.f};
    v8f c1 = {0.f, 0.f, 0.f, 0.f, 0.f, 0.f, 0.f, 0.f};

    const float* __restrict__ xrow = x + (size_t)mrow * NFEAT;
    #pragma unroll 4
    for (int kb = 0; kb < NFEAT; kb += 4) {
        const int k = kb + khalf;
        v2f a;                                    // contiguous float2 of x row
        a.x = xrow[k];
        a.y = xrow[k + 1];
        v2f b0, b1;                               // B 4x16: row K striped on lanes
        b0.x = W1[(size_t)k       * HIDW + ncol0];
        b0.y = W1[(size_t)(k + 1) * HIDW + ncol0];
        b1.x = W1[(size_t)k       * HIDW + ncol1];
        b1.y = W1[(size_t)(k + 1) * HIDW + ncol1];
        c0 = __builtin_amdgcn_wmma_f32_16x16x4_f32(false, a, false, b0,
                                                   (short)0, c0, false, false);
        c1 = __builtin_amdgcn_wmma_f32_16x16x4_f32(false, a, false, b1,
                                                   (short)0, c1, false, false);
    }

    // Store: lane's element r lives at row (tileM + 8*(lane>>4) + r), col ncol{0,1}.
    const int rBase = tileM + ((lane >> 4) * 8);
    float* __restrict__ orow = xw + (size_t)rBase * HIDW;
    if (tileM + 16 <= nrows) {
        // full tile: wave-uniform, unconditional clause-able stores
        #pragma unroll
        for (int r = 0; r < 8; ++r) {
            orow[(size_t)r * HIDW + ncol0] = c0[r];
            orow[(size_t)r * HIDW + ncol1] = c1[r];
        }
    } else {
        #pragma unroll
        for (int r = 0; r < 8; ++r) {
            if (rBase + r < nrows) {
                orow[(size_t)r * HIDW + ncol0] = c0[r];
                orow[(size_t)r * HIDW + ncol1] = c1[r];
            }
        }
    }
}

// ---------------------------------------------------------------------------
// 5) edge scatter: agg[dst] += xw[src] * (dinv[src]*dinv[dst])
//    One wave32 per edge; lane handles 4 contiguous floats (float4 gather +
//    4x global_atomic_add_f32). xw/agg are 51.2MB each -> L2-resident.
// ---------------------------------------------------------------------------
__global__ void __launch_bounds__(256)
gcn_scatter_kernel(const float* __restrict__ xw,
                   const float* __restrict__ dinv,
                   const int* __restrict__ src,
                   const int* __restrict__ dst,
                   float* __restrict__ agg, int nedges) {
    const int gwarp = (int)((blockIdx.x * blockDim.x + threadIdx.x) >> 5);
    const int lane  = threadIdx.x & 31;
    if (gwarp >= nedges) return;

    const int s = src[gwarp];
    const int d = dst[gwarp];
    const float norm = dinv[s] * dinv[d];

    const float4 v = ((const float4*)(xw + (size_t)s * HIDW))[lane];
    float* ar = agg + (size_t)d * HIDW + lane * 4;
    atomicAdd(ar + 0, v.x * norm);
    atomicAdd(ar + 1, v.y * norm);
    atomicAdd(ar + 2, v.z * norm);
    atomicAdd(ar + 3, v.w * norm);
}

// ---------------------------------------------------------------------------
// 6) finalize: h = relu(agg + xw*dinv^2 + b1);  out = [x,h] @ Wfc + bfc
//    One wave32 per node; lane covers 4 features of each half; Wfc is 2KB
//    (cache resident). Wave shuffle reduction, lane 0 writes 2 outputs.
// ---------------------------------------------------------------------------
__global__ void __launch_bounds__(256)
gcn_finalize_kernel(const float* __restrict__ x,
                    const float* __restrict__ xw,
                    const float* __restrict__ agg,
                    const float* __restrict__ dinv,
                    const float* __restrict__ b1,
                    const float* __restrict__ Wfc,
                    const float* __restrict__ bfc,
                    float* __restrict__ out, int n) {
    const int node = (int)((blockIdx.x * blockDim.x + threadIdx.x) >> 5);
    const int lane = threadIdx.x & 31;
    if (node >= n) return;

    const float di   = dinv[node];
    const float self = di * di;

    const float4 xv = ((const float4*)(x   + (size_t)node * NFEAT))[lane];
    const float4 wv = ((const float4*)(xw  + (size_t)node * HIDW))[lane];
    const float4 av = ((const float4*)(agg + (size_t)node * HIDW))[lane];
    const float4 bv = ((const float4*)b1)[lane];

    float4 h;
    h.x = fmaxf(av.x + wv.x * self + bv.x, 0.0f);
    h.y = fmaxf(av.y + wv.y * self + bv.y, 0.0f);
    h.z = fmaxf(av.z + wv.z * self + bv.z, 0.0f);
    h.w = fmaxf(av.w + wv.w * self + bv.w, 0.0f);

    const int f0 = lane * 4;              // x-part rows f0..f0+3; h-part +128
    float acc0, acc1;
    acc0  = xv.x * Wfc[(f0 + 0) * 2 + 0] + xv.y * Wfc[(f0 + 1) * 2 + 0]
          + xv.z * Wfc[(f0 + 2) * 2 + 0] + xv.w * Wfc[(f0 + 3) * 2 + 0];
    acc1  = xv.x * Wfc[(f0 + 0) * 2 + 1] + xv.y * Wfc[(f0 + 1) * 2 + 1]
          + xv.z * Wfc[(f0 + 2) * 2 + 1] + xv.w * Wfc[(f0 + 3) * 2 + 1];
    acc0 += h.x * Wfc[(NFEAT + f0 + 0) * 2 + 0] + h.y * Wfc[(NFEAT + f0 + 1) * 2 + 0]
          + h.z * Wfc[(NFEAT + f0 + 2) * 2 + 0] + h.w * Wfc[(NFEAT + f0 + 3) * 2 + 0];
    acc1 += h.x * Wfc[(NFEAT + f0 + 0) * 2 + 1] + h.y * Wfc[(NFEAT + f0 + 1) * 2 + 1]
          + h.z * Wfc[(NFEAT + f0 + 2) * 2 + 1] + h.w * Wfc[(NFEAT + f0 + 3) * 2 + 1];

    #pragma unroll
    for (int off = 16; off > 0; off >>= 1) {
        acc0 += __shfl_down(acc0, off, 32);
        acc1 += __shfl_down(acc1, off, 32);
    }
    if (lane == 0) {
        out[(size_t)node * 2 + 0] = acc0 + bfc[0];
        out[(size_t)node * 2 + 1] = acc1 + bfc[1];
    }
}

// ---------------------------------------------------------------------------
// launcher
// ---------------------------------------------------------------------------
extern "C" void kernel_launch(void* const* d_in, const int* in_sizes, int n_in,
                              void* d_out, int out_size, void* d_ws, size_t ws_size,
                              hipStream_t stream) {
    const float* x   = (const float*)d_in[0];
    const int*   ei  = (const int*)  d_in[1];   // [2, E] flat: src then dst
    const float* W1  = (const float*)d_in[2];
    const float* b1  = (const float*)d_in[3];
    const float* Wfc = (const float*)d_in[4];
    const float* bfc = (const float*)d_in[5];
    float*       out = (float*)d_out;

    const int N = in_sizes[0] / NFEAT;          // 100000
    const int E = in_sizes[1] / 2;              // 1600000
    const int* src = ei;
    const int* dst = ei + E;

    // workspace carve-up: dinv[N] | xw[N*128] | agg[N*128]
    float* deg = (float*)d_ws;                  // becomes dinv in-place
    float* xw  = deg + N;
    float* agg = xw + (size_t)N * HIDW;

    const int totalAgg = N * HIDW;

    // 1) init (must run every call: harness poisons ws once, graph replays)
    gcn_init_kernel<<<(totalAgg + 255) / 256, 256, 0, stream>>>(deg, agg, N, totalAgg);

    // 2) in-degree via f32 atomics
    gcn_degree_kernel<<<(E + 255) / 256, 256, 0, stream>>>(dst, deg, E);

    // 3) dinv = rsqrt(deg)
    gcn_rsqrt_kernel<<<(N + 255) / 256, 256, 0, stream>>>(deg, N);

    // 4) xw = x @ W1  (WMMA fp32)
    gcn_gemm_xw_kernel<<<(N + 15) / 16, 128, 0, stream>>>(x, W1, xw, N);

    // 5) edge gather/scale/scatter-add (1 wave per edge)
    {
        const long long threads = (long long)E * 32;
        const int blocks = (int)((threads + 255) / 256);
        gcn_scatter_kernel<<<blocks, 256, 0, stream>>>(xw, deg, src, dst, agg, E);
    }

    // 6) relu + self-loop + fused concat-GEMV (1 wave per node)
    {
        const long long threads = (long long)N * 32;
        const int blocks = (int)((threads + 255) / 256);
        gcn_finalize_kernel<<<blocks, 256, 0, stream>>>(x, xw, agg, deg, b1, Wfc, bfc, out, N);
    }
}